// Filter_poly_22926535426502
// MI455X (gfx1250) — compile-verified
//
#include <hip/hip_runtime.h>
#include <math.h>

typedef __attribute__((ext_vector_type(2))) float v2f;
typedef __attribute__((ext_vector_type(8))) float v8f;

#define T_LEN 16384
#define B_LEN 256
#define NSLICE 32
#define KS (T_LEN / NSLICE)     // 512 time steps per K-slice
#define NBTILE 16               // 256 / 16
#define CHUNKS 64
#define CLEN (T_LEN / CHUNKS)   // 256 steps per scan chunk

// workspace layout (float offsets)
#define WS_PART 0                                    // [NSLICE][NBTILE][6][16]
#define WS_G    (WS_PART + NSLICE * NBTILE * 6 * 16) // [6][256]
#define WS_ROWP (WS_G + 6 * 256)                     // [256][8]
#define WS_CM   (WS_ROWP + 256 * 8)                  // [256][CHUNKS][10]

struct InvMat { float v[36]; };

// ---------------------------------------------------------------------------
// Kernel 1: G_partial = A^T @ Y via V_WMMA_F32_16X16X4_F32.
// Y[t][b] = Xflat[t*256 + b]  (the reshape view). A[t][j] = (t/T)^(5-j).
// M = poly feature (rows 6..15 forced to 0), N = 16 columns of b, K = 4 t's.
// One wave per (kslice, btile); 512 waves; fp32 partials to ws.
// Power selection is branchless: one-hot lane weights + Horner -> 5 FMA/elem.
// ---------------------------------------------------------------------------
__global__ void k_poly_partials(const float* __restrict__ Xf,
                                float* __restrict__ part) {
  const int lane   = threadIdx.x & 31;
  const int wave   = blockIdx.x * 8 + (threadIdx.x >> 5); // 8 waves / block
  const int btile  = wave & (NBTILE - 1);
  const int kslice = wave >> 4;
  const int m      = lane & 15;   // A: M index, B: N index
  const int khalf  = lane >> 4;   // selects K pair (ISA 16x4 f32 layout)
  const float invT = 1.0f / (float)T_LEN;

  // one-hot weights (loop invariant): p(x) = sum_e w_e * x^e = x^(5-m)
  const float w5 = (m == 0) ? 1.f : 0.f;
  const float w4 = (m == 1) ? 1.f : 0.f;
  const float w3 = (m == 2) ? 1.f : 0.f;
  const float w2 = (m == 3) ? 1.f : 0.f;
  const float w1 = (m == 4) ? 1.f : 0.f;
  const float w0 = (m == 5) ? 1.f : 0.f;

  v8f c = {0.f, 0.f, 0.f, 0.f, 0.f, 0.f, 0.f, 0.f};
  const int t_begin = kslice * KS;

  // x for this lane's two K elements; advances exactly by 4/T per iteration
  // (all values are multiples of 2^-14 < 1 -> fp32-exact increments).
  float xA = (float)(t_begin + 2 * khalf) * invT;
  float xB = xA + invT;
  const float xstep = 4.0f * invT;

  const float* p = Xf + (size_t)(t_begin + 2 * khalf) * B_LEN
                      + (size_t)btile * 16 + m;

#pragma unroll 4
  for (int it = 0; it < KS / 4; ++it) {
    v2f a, b;
    a.x = fmaf(fmaf(fmaf(fmaf(fmaf(w5, xA, w4), xA, w3), xA, w2), xA, w1), xA, w0);
    a.y = fmaf(fmaf(fmaf(fmaf(fmaf(w5, xB, w4), xB, w3), xB, w2), xB, w1), xB, w0);
    b.x = p[0];       // Y[tk  ][btile*16 + n]
    b.y = p[B_LEN];   // Y[tk+1][btile*16 + n]
    c = __builtin_amdgcn_wmma_f32_16x16x4_f32(false, a, false, b,
                                              (short)0, c, false, false);
    xA += xstep;
    xB += xstep;
    p += 4 * B_LEN;
  }
  if (lane < 16) {   // C/D: lanes 0-15 hold M = vgpr index (we need M=0..5)
    float* dst = part + (((size_t)kslice * NBTILE + btile) * 6) * 16 + lane;
#pragma unroll
    for (int r = 0; r < 6; ++r) dst[r * 16] = c[r];
  }
}

// ---------------------------------------------------------------------------
// Kernel 2: fixed-order slice reduction (deterministic, no float atomics).
// ---------------------------------------------------------------------------
__global__ void k_reduce_G(const float* __restrict__ part,
                           float* __restrict__ G) {
  const int id = blockIdx.x * blockDim.x + threadIdx.x;
  if (id >= 6 * 256) return;
  const int j = id >> 8, b = id & 255;
  const int btile = b >> 4, n = b & 15;
  float s = 0.f;
  for (int sl = 0; sl < NSLICE; ++sl)
    s += part[(((size_t)sl * NBTILE + btile) * 6 + j) * 16 + n];
  G[j * 256 + b] = s;
}

// ---------------------------------------------------------------------------
// Kernel 3: coeffs = AtA^-1 @ G, BN -> MLP -> BN -> relu -> MLP -> sigmoid,
// then per-row filter constants. One block of 256 threads (thread = row b).
// ---------------------------------------------------------------------------
__global__ void k_mlp_params(const float* __restrict__ G,
                             const float* __restrict__ bn1g, const float* __restrict__ bn1b,
                             const float* __restrict__ W1,   const float* __restrict__ b1,
                             const float* __restrict__ bn2g, const float* __restrict__ bn2b,
                             const float* __restrict__ W2,   const float* __restrict__ b2,
                             const float* __restrict__ pn,
                             const InvMat inv,
                             float* __restrict__ rowpar) {
  __shared__ float s1[6][256];
  __shared__ float s2[8][256];
  __shared__ float mv1[6][2];
  __shared__ float mv2[8][2];
  const int b = threadIdx.x;

  float feat[6];
#pragma unroll
  for (int j = 0; j < 6; ++j) {
    float s = 0.f;
#pragma unroll
    for (int i = 0; i < 6; ++i) s += inv.v[j * 6 + i] * G[i * 256 + b];
    feat[j] = s;
    s1[j][b] = s;
  }
  __syncthreads();
  if (b < 6) {
    float m = 0.f;
    for (int i = 0; i < 256; ++i) m += s1[b][i];
    m *= (1.0f / 256.0f);
    float v = 0.f;
    for (int i = 0; i < 256; ++i) { float d = s1[b][i] - m; v += d * d; }
    v *= (1.0f / 256.0f);
    mv1[b][0] = m; mv1[b][1] = v;
  }
  __syncthreads();

  float hat[6];
#pragma unroll
  for (int j = 0; j < 6; ++j)
    hat[j] = (feat[j] - mv1[j][0]) / sqrtf(mv1[j][1] + 1e-5f) * bn1g[j] + bn1b[j];

  float h[8];
#pragma unroll
  for (int o = 0; o < 8; ++o) {
    float s = b1[o];
#pragma unroll
    for (int j = 0; j < 6; ++j) s += hat[j] * W1[o * 6 + j];
    h[o] = s;
    s2[o][b] = s;
  }
  __syncthreads();
  if (b < 8) {
    float m = 0.f;
    for (int i = 0; i < 256; ++i) m += s2[b][i];
    m *= (1.0f / 256.0f);
    float v = 0.f;
    for (int i = 0; i < 256; ++i) { float d = s2[b][i] - m; v += d * d; }
    v *= (1.0f / 256.0f);
    mv2[b][0] = m; mv2[b][1] = v;
  }
  __syncthreads();

  float hr[8];
#pragma unroll
  for (int o = 0; o < 8; ++o) {
    float z = (h[o] - mv2[o][0]) / sqrtf(mv2[o][1] + 1e-5f) * bn2g[o] + bn2b[o];
    hr[o] = z > 0.f ? z : 0.f;
  }

  const float lo[6] = {100.f, 5000.f, 0.5f, 0.01f, 0.f, 0.001f};
  const float hi[6] = {5000.f, 20000.f, 0.999f, 1.f, 1.f, 0.1f};
  float p[6];
#pragma unroll
  for (int j = 0; j < 6; ++j) {
    float s = b2[j];
#pragma unroll
    for (int o = 0; o < 8; ++o) s += hr[o] * W2[j * 8 + o];
    s += pn[j];
    float sg = 1.f / (1.f + expf(-s));
    p[j] = lo[j] + (hi[j] - lo[j]) * sg;
  }
  // p: f_inf, f_start, f_decay, f_T, w_offset, w_T
  float* rp = rowpar + b * 8;
  rp[0] = p[0];                           // f_inf
  rp[1] = p[1] - p[0];                    // f_start - f_inf
  rp[2] = log2f(p[2]) / p[3];             // k1: f = f_inf + df*exp2(k1*tn)
  rp[3] = p[4];                           // w_offset
  rp[4] = 1.4426950408889634f / p[5];     // k2l: w = sigmoid via exp2
}

// ---------------------------------------------------------------------------
// Kernel 4: chunked parallel scan. Step map on (y1,y2,acc) is affine &
// lower-triangular; each thread composes CLEN step maps for one (row,chunk).
// ---------------------------------------------------------------------------
__global__ void k_chunk_scan(const float* __restrict__ Xf,
                             const float* __restrict__ rowpar,
                             float* __restrict__ cm, const float Cf) {
  const int g = blockIdx.x * blockDim.x + threadIdx.x; // 0..16383
  const int row = g >> 6;
  const int chunk = g & (CHUNKS - 1);
  const float* rp = rowpar + row * 8;
  const float f_inf = rp[0], df = rp[1], k1 = rp[2], woff = rp[3], k2l = rp[4];
  const float invT = 1.0f / (float)T_LEN;
  const float* Xr = Xf + (size_t)row * T_LEN;

  const int ts = chunk * CLEN;
  const int te = ts + CLEN;
  int t = (chunk == 0) ? 1 : ts;          // scan starts at t = 1
  float xp = Xr[t - 1];
  float P = 1.f, Q = 0.f, R = 0.f, S = 1.f, U = 0.f;
  float V = 0.f, W = 0.f, Z = 0.f, wsum = 0.f;

  for (; t < te; ++t) {
    const float xt = Xr[t];
    const float tn = (float)t * invT;
    const float f  = fmaf(df, exp2f(k1 * tn), f_inf);
    const float a  = (f - Cf) / (f + Cf);
    const float bb = 0.5f * (a + 1.f);
    const float w  = 1.f / (1.f + exp2f(-k2l * (tn - woff)));
    // step: y1' = alpha*y1 + beta ; y2' = gamma*y1 + alpha*y2 + delta
    const float alpha = -a;
    const float beta  = bb * (xt + xp);
    const float gamma = bb * (1.f + alpha);
    const float delta = bb * beta;
    // compose onto accumulated map {P,Q | R,S,U | V,W,Z}
    const float Rn = fmaf(gamma, P, alpha * R);
    const float Un = fmaf(gamma, Q, fmaf(alpha, U, delta));
    const float Sn = alpha * S;
    const float Pn = alpha * P;
    const float Qn = fmaf(alpha, Q, beta);
    V = fmaf(w, Rn, V); W = fmaf(w, Sn, W); Z = fmaf(w, Un, Z);
    wsum += w;
    P = Pn; Q = Qn; R = Rn; S = Sn; U = Un;
    xp = xt;
  }
  float* dst = cm + ((size_t)row * CHUNKS + chunk) * 10;
  dst[0] = P; dst[1] = Q; dst[2] = R; dst[3] = S; dst[4] = U;
  dst[5] = V; dst[6] = W; dst[7] = Z; dst[8] = wsum;
}

// ---------------------------------------------------------------------------
// Kernel 5: sequential combine of 64 chunk maps per row (trivial), output.
// ---------------------------------------------------------------------------
__global__ void k_combine(const float* __restrict__ Xf,
                          const float* __restrict__ rowpar,
                          const float* __restrict__ cm,
                          float* __restrict__ out) {
  const int row = threadIdx.x;
  const float* rp = rowpar + row * 8;
  const float woff = rp[3], k2l = rp[4];
  const float x0 = Xf[(size_t)row * T_LEN];
  float y1 = x0, y2 = x0, acc = 0.f, norm = 0.f;
  const float* c = cm + (size_t)row * CHUNKS * 10;
  for (int k = 0; k < CHUNKS; ++k, c += 10) {
    const float P = c[0], Q = c[1], R = c[2], S = c[3], U = c[4];
    const float V = c[5], W = c[6], Z = c[7], ws = c[8];
    acc = acc + V * y1 + W * y2 + Z;
    const float y1n = fmaf(P, y1, Q);
    const float y2n = fmaf(R, y1, fmaf(S, y2, U));
    y1 = y1n; y2 = y2n;
    norm += ws;
  }
  const float w0 = 1.f / (1.f + exp2f(k2l * woff)); // sigmoid((0-woff)/wT)
  out[row] = fmaf(w0, x0, acc) / (norm + w0);
}

// ---------------------------------------------------------------------------
extern "C" void kernel_launch(void* const* d_in, const int* in_sizes, int n_in,
                              void* d_out, int out_size, void* d_ws, size_t ws_size,
                              hipStream_t stream) {
  (void)in_sizes; (void)n_in; (void)out_size; (void)ws_size;
  const float* X    = (const float*)d_in[0];
  const float* bn1g = (const float*)d_in[1];
  const float* bn1b = (const float*)d_in[2];
  const float* W1   = (const float*)d_in[3];
  const float* b1   = (const float*)d_in[4];
  const float* bn2g = (const float*)d_in[5];
  const float* bn2b = (const float*)d_in[6];
  const float* W2   = (const float*)d_in[7];
  const float* b2   = (const float*)d_in[8];
  const float* pn   = (const float*)d_in[9];

  float* ws     = (float*)d_ws;
  float* part   = ws + WS_PART;
  float* G      = ws + WS_G;
  float* rowpar = ws + WS_ROWP;
  float* cmw    = ws + WS_CM;

  // Host-side constant prep (pure function of T -> deterministic,
  // graph-capture safe): AtA power sums in double, entries rounded to f32,
  // f32 Gauss-Jordan inverse with partial pivoting (mirrors f32 LU solve).
  double Sp[11];
  for (int p = 0; p <= 10; ++p) Sp[p] = 0.0;
  for (int t = 0; t < T_LEN; ++t) {
    const double x = (double)t / (double)T_LEN;
    double xp = 1.0;
    for (int p = 0; p <= 10; ++p) { Sp[p] += xp; xp *= x; }
  }
  float Af[6][12];
  for (int i = 0; i < 6; ++i) {
    for (int j = 0; j < 6; ++j) Af[i][j] = (float)Sp[10 - i - j];
    for (int j = 0; j < 6; ++j) Af[i][6 + j] = (i == j) ? 1.f : 0.f;
  }
  for (int col = 0; col < 6; ++col) {
    int piv = col;
    float best = fabsf(Af[col][col]);
    for (int r = col + 1; r < 6; ++r) {
      float v = fabsf(Af[r][col]);
      if (v > best) { best = v; piv = r; }
    }
    if (piv != col)
      for (int k = 0; k < 12; ++k) {
        float tmp = Af[col][k]; Af[col][k] = Af[piv][k]; Af[piv][k] = tmp;
      }
    const float d = 1.0f / Af[col][col];
    for (int k = 0; k < 12; ++k) Af[col][k] *= d;
    for (int r = 0; r < 6; ++r) {
      if (r == col) continue;
      const float fct = Af[r][col];
      for (int k = 0; k < 12; ++k) Af[r][k] -= fct * Af[col][k];
    }
  }
  InvMat inv;
  for (int i = 0; i < 6; ++i)
    for (int j = 0; j < 6; ++j) inv.v[i * 6 + j] = Af[i][6 + j];

  const float Cf =
      (float)(sqrt(cbrt(2.0) - 1.0) * 50000.0 / 3.14159265358979323846);

  k_poly_partials<<<64, 256, 0, stream>>>(X, part);
  k_reduce_G<<<6, 256, 0, stream>>>(part, G);
  k_mlp_params<<<1, 256, 0, stream>>>(G, bn1g, bn1b, W1, b1, bn2g, bn2b,
                                      W2, b2, pn, inv, rowpar);
  k_chunk_scan<<<64, 256, 0, stream>>>(X, rowpar, cmw, Cf);
  k_combine<<<1, 256, 0, stream>>>(X, rowpar, cmw, (float*)d_out);
}